// gnn_v3_53652731461899
// MI455X (gfx1250) — compile-verified
//
#include <hip/hip_runtime.h>
#include <hip/hip_bf16.h>

typedef float v2f __attribute__((ext_vector_type(2)));
typedef float v8f __attribute__((ext_vector_type(8)));

static __device__ __forceinline__ v8f wmma4(v2f a, v2f b, v8f c) {
  // D = A(16x4 f32) * B(4x16 f32) + C(16x16 f32), full fp32 precision
  return __builtin_amdgcn_wmma_f32_16x16x4_f32(false, a, false, b, (short)0, c,
                                               false, false);
}

static __device__ __forceinline__ v8f splat8(float v) {
  v8f r;
#pragma unroll
  for (int i = 0; i < 8; ++i) r[i] = v;
  return r;
}

// ---------------------------------------------------------------------------
// x_next = x @ root + bias     (N rows; 64x64 weight transposed in LDS)
// 256 threads = 8 waves, each wave owns one 16-row tile.
// ---------------------------------------------------------------------------
__global__ __launch_bounds__(256) void k_root(const float* __restrict__ x,
                                              const float* __restrict__ W,
                                              const float* __restrict__ b,
                                              float* __restrict__ xn,
                                              int ntiles) {
  __shared__ float Ws[64 * 66];  // transposed [n][k], pad 66 for b64 loads
  const int t = threadIdx.x;
#pragma unroll
  for (int q = 0; q < 4; ++q) {
    const int idx = q * 1024 + t * 4;
    const float4 v = *(const float4*)&W[idx];
    const int k = idx >> 6, n0 = idx & 63;
    Ws[(n0 + 0) * 66 + k] = v.x;
    Ws[(n0 + 1) * 66 + k] = v.y;
    Ws[(n0 + 2) * 66 + k] = v.z;
    Ws[(n0 + 3) * 66 + k] = v.w;
  }
  __syncthreads();
  const int wave = t >> 5, lane = t & 31;
  const int m = lane & 15, lg = lane >> 4;
  const int traw = blockIdx.x * 8 + wave;
  const int tile = traw < ntiles ? traw : ntiles - 1;
  const bool act = traw < ntiles;
  const int r0 = tile * 16;

  v8f acc[4];
#pragma unroll
  for (int nt = 0; nt < 4; ++nt) acc[nt] = splat8(b[nt * 16 + m]);

#pragma unroll
  for (int kb = 0; kb < 16; ++kb) {
    const int kk = kb * 4 + 2 * lg;
    v2f a = *(const v2f*)&x[(r0 + m) * 64 + kk];
#pragma unroll
    for (int nt = 0; nt < 4; ++nt) {
      v2f bb = *(const v2f*)&Ws[(nt * 16 + m) * 66 + kk];  // ds_load_b64
      acc[nt] = wmma4(a, bb, acc[nt]);
    }
  }
  if (act) {
#pragma unroll
    for (int nt = 0; nt < 4; ++nt)
#pragma unroll
      for (int v = 0; v < 8; ++v)
        xn[(r0 + v + 8 * lg) * 64 + nt * 16 + m] = acc[nt][v];
  }
}

// ---------------------------------------------------------------------------
// Edge kernel network: h2 = relu(relu(e@W1+b1)@W2+b2)  ->  hout [E,64]
// 256 threads = 8 waves, each wave owns one 16-edge tile.
// W1 (transposed, pad 10) and W2 (transposed, pad 66) staged in LDS.
// ---------------------------------------------------------------------------
__global__ __launch_bounds__(256) void k_edge_mlp(
    const float* __restrict__ ef, const float* __restrict__ W1,
    const float* __restrict__ b1, const float* __restrict__ W2,
    const float* __restrict__ b2, float* __restrict__ hout, int etiles) {
  __shared__ float W1s[64 * 10];       // 2.5 KB transposed [n][k], pad 10
  __shared__ float W2s[64 * 66];       // 16.9 KB transposed [n][k], pad 66
  __shared__ float h1s[8 * 16 * 64];   // 32 KB, per-wave 16x64 tile
  const int t = threadIdx.x;
  {  // stage W1 transposed: 512 floats, 2 per thread
    const int idx = t * 2;
    const v2f v = *(const v2f*)&W1[idx];
    const int k = idx >> 6, n0 = idx & 63;
    W1s[(n0 + 0) * 10 + k] = v.x;
    W1s[(n0 + 1) * 10 + k] = v.y;
  }
#pragma unroll
  for (int q = 0; q < 4; ++q) {  // stage W2 transposed
    const int idx = q * 1024 + t * 4;
    const float4 v = *(const float4*)&W2[idx];
    const int k = idx >> 6, n0 = idx & 63;
    W2s[(n0 + 0) * 66 + k] = v.x;
    W2s[(n0 + 1) * 66 + k] = v.y;
    W2s[(n0 + 2) * 66 + k] = v.z;
    W2s[(n0 + 3) * 66 + k] = v.w;
  }
  __syncthreads();
  const int wave = t >> 5, lane = t & 31;
  const int m = lane & 15, lg = lane >> 4;
  const int traw = blockIdx.x * 8 + wave;
  const int tile = traw < etiles ? traw : etiles - 1;
  const bool act = traw < etiles;
  const int e0 = tile * 16;
  float* myh1 = &h1s[wave * 1024];

  // GEMM1: [16,8] @ [8,64], K = 8 -> 2 K-blocks
  v8f a1[4];
#pragma unroll
  for (int nt = 0; nt < 4; ++nt) a1[nt] = splat8(b1[nt * 16 + m]);
#pragma unroll
  for (int kb = 0; kb < 2; ++kb) {
    const int kk = kb * 4 + 2 * lg;
    v2f a = *(const v2f*)&ef[(e0 + m) * 8 + kk];
#pragma unroll
    for (int nt = 0; nt < 4; ++nt) {
      v2f bb = *(const v2f*)&W1s[(nt * 16 + m) * 10 + kk];  // ds_load_b64
      a1[nt] = wmma4(a, bb, a1[nt]);
    }
  }
  // relu -> LDS (D layout rows v + 8*lg)
#pragma unroll
  for (int nt = 0; nt < 4; ++nt)
#pragma unroll
    for (int v = 0; v < 8; ++v)
      myh1[(v + 8 * lg) * 64 + nt * 16 + m] = fmaxf(a1[nt][v], 0.0f);
  __syncthreads();  // also orders LDS within wave

  // GEMM2: [16,64] @ [64,64]
  v8f a2[4];
#pragma unroll
  for (int nt = 0; nt < 4; ++nt) a2[nt] = splat8(b2[nt * 16 + m]);
#pragma unroll
  for (int kb = 0; kb < 16; ++kb) {
    const int kk = kb * 4 + 2 * lg;
    v2f a = *(const v2f*)&myh1[m * 64 + kk];
#pragma unroll
    for (int nt = 0; nt < 4; ++nt) {
      v2f bb = *(const v2f*)&W2s[(nt * 16 + m) * 66 + kk];  // ds_load_b64
      a2[nt] = wmma4(a, bb, a2[nt]);
    }
  }
  if (act) {
#pragma unroll
    for (int nt = 0; nt < 4; ++nt)
#pragma unroll
      for (int v = 0; v < 8; ++v)
        hout[(e0 + v + 8 * lg) * 64 + nt * 16 + m] = fmaxf(a2[nt][v], 0.0f);
  }
}

// ---------------------------------------------------------------------------
// Fused message GEMM + scatter:  msg = P @ W3r,  P[e,(c,i)] = h2[e,c]*xs[e,i]
// (+ one extra K-row block of b3 with h==1), atomic scatter-add into xn.
// 128 threads = 4 waves; block owns 64 edges; xs/h2 tiles + one transposed
// W3 chunk (64x66 pad) live in LDS.  65 K-chunks x 16 ib x 4 nt WMMA.
// ---------------------------------------------------------------------------
__global__ __launch_bounds__(128) void k_msg(
    const float* __restrict__ xcur, const float* __restrict__ hbuf,
    const float* __restrict__ W3, const float* __restrict__ b3,
    const int* __restrict__ src, const int* __restrict__ tgt,
    float* __restrict__ xn, int etiles, int E) {
  __shared__ float xss[64 * 64];   // 16 KB  gathered x[src]
  __shared__ float h2s[64 * 64];   // 16 KB  h2 tile
  __shared__ float w3s[64 * 66];   // 16.9 KB transposed chunk [o][i], pad 66
  const int t = threadIdx.x;                   // 0..127
  const int wave = t >> 5, lane = t & 31;
  const int m = lane & 15, lg = lane >> 4;
  const int tile0 = blockIdx.x * 4;
  const int traw = tile0 + wave;
  const int tile = traw < etiles ? traw : etiles - 1;
  const bool act = traw < etiles;

  // cooperative gather of xs and h2 tiles: thread t -> row t/2, half t&1
  {
    const int row = t >> 1, half = t & 1;
    int eIdx = tile0 * 16 + row;
    if (eIdx >= E) eIdx = E - 1;
    const int s = src[eIdx];
    const float* xr = &xcur[(size_t)s * 64 + half * 32];
    const float* hr = &hbuf[(size_t)eIdx * 64 + half * 32];
    float* xd = &xss[row * 64 + half * 32];
    float* hd = &h2s[row * 64 + half * 32];
#pragma unroll
    for (int q = 0; q < 8; ++q) *(float4*)&xd[q * 4] = *(const float4*)&xr[q * 4];
#pragma unroll
    for (int q = 0; q < 8; ++q) *(float4*)&hd[q * 4] = *(const float4*)&hr[q * 4];
  }
  __syncthreads();

  v8f acc[4];
#pragma unroll
  for (int nt = 0; nt < 4; ++nt) acc[nt] = splat8(0.0f);

  const int lrow = (tile - tile0) * 16 + m;  // row inside LDS tiles

  for (int c = 0; c < 65; ++c) {
    __syncthreads();  // previous chunk fully consumed
    {                 // stage chunk c (W3 row c, or b3 for c==64), transposed
      const float* srow = (c < 64) ? (W3 + (size_t)c * 4096) : b3;
      const int i = t >> 1, ob = (t & 1) * 32;
      const float* sp = &srow[i * 64 + ob];
#pragma unroll
      for (int q = 0; q < 8; ++q) {
        float4 v = *(const float4*)&sp[q * 4];
        const int o = ob + q * 4;
        w3s[(o + 0) * 66 + i] = v.x;
        w3s[(o + 1) * 66 + i] = v.y;
        w3s[(o + 2) * 66 + i] = v.z;
        w3s[(o + 3) * 66 + i] = v.w;
      }
    }
    __syncthreads();

    const float hc = (c < 64) ? h2s[lrow * 64 + c] : 1.0f;
#pragma unroll
    for (int ib = 0; ib < 16; ++ib) {
      const int kk = ib * 4 + 2 * lg;
      v2f xv = *(const v2f*)&xss[lrow * 64 + kk];
      v2f av;
      av.x = hc * xv.x;
      av.y = hc * xv.y;
#pragma unroll
      for (int nt = 0; nt < 4; ++nt) {
        v2f bv = *(const v2f*)&w3s[(nt * 16 + m) * 66 + kk];  // ds_load_b64
        acc[nt] = wmma4(av, bv, acc[nt]);
      }
    }
  }

  if (act) {
    const int e0 = tile * 16;
#pragma unroll
    for (int v = 0; v < 8; ++v) {
      const int tg = tgt[e0 + v + 8 * lg];
#pragma unroll
      for (int nt = 0; nt < 4; ++nt)
        atomicAdd(&xn[(size_t)tg * 64 + nt * 16 + m], acc[nt][v]);
    }
  }
}

// ---------------------------------------------------------------------------
// Global pooling + tiny head
// ---------------------------------------------------------------------------
__global__ void k_zero(float* __restrict__ g) { g[threadIdx.x] = 0.0f; }

__global__ __launch_bounds__(256) void k_pool(const float* __restrict__ x,
                                              float* __restrict__ g, int N) {
  const int t = threadIdx.x;
  const int col = t & 63;
  const int rseg = t >> 6;  // 0..3
  float s = 0.0f;
  for (int r = blockIdx.x * 4 + rseg; r < N; r += gridDim.x * 4)
    s += x[(size_t)r * 64 + col];
  atomicAdd(&g[col], s);
}

__global__ void k_final(const float* __restrict__ g,
                        const float* __restrict__ d1w,
                        const float* __restrict__ d1b,
                        const float* __restrict__ d2w,
                        const float* __restrict__ d2b,
                        float* __restrict__ out) {
  __shared__ float red[64];
  const int t = threadIdx.x;  // 64 threads
  float acc = d1b[t];
#pragma unroll
  for (int k = 0; k < 64; ++k) acc += g[k] * d1w[k * 64 + t];
  acc = fmaxf(acc, 0.0f);
  red[t] = acc * d2w[t];
  __syncthreads();
  if (t == 0) {
    float s = d2b[0];
#pragma unroll
    for (int k = 0; k < 64; ++k) s += red[k];
    out[0] = s;
  }
}

// ---------------------------------------------------------------------------
extern "C" void kernel_launch(void* const* d_in, const int* in_sizes, int n_in,
                              void* d_out, int out_size, void* d_ws,
                              size_t ws_size, hipStream_t stream) {
  const float* x0 = (const float*)d_in[0];
  const int* src = (const int*)d_in[1];
  const int* tgt = (const int*)d_in[2];
  const float* ef = (const float*)d_in[3];
  const float* w1 = (const float*)d_in[4];
  const float* b1 = (const float*)d_in[5];
  const float* w2 = (const float*)d_in[6];
  const float* b2 = (const float*)d_in[7];
  const float* w3 = (const float*)d_in[8];
  const float* b3 = (const float*)d_in[9];
  const float* root = (const float*)d_in[10];
  const float* bias = (const float*)d_in[11];
  const float* d1w = (const float*)d_in[12];
  const float* d1b = (const float*)d_in[13];
  const float* d2w = (const float*)d_in[14];
  const float* d2b = (const float*)d_in[15];

  const int N = in_sizes[0] / 64;
  const int E = in_sizes[1];
  const int L = in_sizes[4] / (8 * 64);

  const int ntiles = (N + 15) / 16;   // 625
  const int etiles = (E + 15) / 16;   // 3125

  float* xa = (float*)d_ws;
  float* xb = xa + (size_t)N * 64;
  float* hh = xb + (size_t)N * 64;
  float* gg = hh + (size_t)E * 64;

  const float* xin = x0;
  float* xout = xa;
  for (int l = 0; l < L; ++l) {
    xout = (l & 1) ? xb : xa;
    k_root<<<(ntiles + 7) / 8, 256, 0, stream>>>(
        xin, root + (size_t)l * 4096, bias + (size_t)l * 64, xout, ntiles);
    k_edge_mlp<<<(etiles + 7) / 8, 256, 0, stream>>>(
        ef, w1 + (size_t)l * 512, b1 + (size_t)l * 64, w2 + (size_t)l * 4096,
        b2 + (size_t)l * 64, hh, etiles);
    k_msg<<<(etiles + 3) / 4, 128, 0, stream>>>(
        xin, hh, w3 + (size_t)l * 64 * 4096, b3 + (size_t)l * 4096, src, tgt,
        xout, etiles, E);
    xin = xout;
  }

  k_zero<<<1, 64, 0, stream>>>(gg);
  k_pool<<<64, 256, 0, stream>>>(xin, gg, N);
  k_final<<<1, 64, 0, stream>>>(gg, d1w, d1b, d2w, d2b, (float*)d_out);
}